// DilatedOCA_48730698940717
// MI455X (gfx1250) — compile-verified
//
#include <hip/hip_runtime.h>
#include <hip/hip_bf16.h>

typedef __attribute__((ext_vector_type(2))) float v2f;
typedef __attribute__((ext_vector_type(8))) float v8f;

__device__ __forceinline__ v8f wmma4(v2f a, v2f b, v8f c) {
  // D = A(16x4 f32) * B(4x16 f32) + C(16x16 f32), full fp32 matrix op
  return __builtin_amdgcn_wmma_f32_16x16x4_f32(false, a, false, b, (short)0, c,
                                               false, false);
}

__device__ __forceinline__ float swap16(float x) {
  // ds_swizzle group-of-32, xor_mask=0x10: exchange lanes l <-> l^16
  int i = __builtin_amdgcn_ds_swizzle(__float_as_int(x), 0x401F);
  return __int_as_float(i);
}

// ---------------------------------------------------------------------------
// Kernel 1: RMS-ish norm (divide by sqrt(var), mean NOT subtracted) per pixel.
// x: [c=64][4096] -> XN: [pix=4096][c=64]
// ---------------------------------------------------------------------------
__global__ __launch_bounds__(256) void norm_kernel(const float* __restrict__ x,
                                                   const float* __restrict__ lnw,
                                                   float* __restrict__ xn) {
  int pix = blockIdx.x * 256 + threadIdx.x;  // 4096 pixels
  float vals[64];
  float s = 0.f, ss = 0.f;
#pragma unroll
  for (int c = 0; c < 64; ++c) {
    float v = x[c * 4096 + pix];
    vals[c] = v;
    s += v;
    ss += v * v;
  }
  float mean = s * (1.f / 64.f);
  float var = ss * (1.f / 64.f) - mean * mean;
  float inv = 1.f / __builtin_sqrtf(var + 1e-5f);
#pragma unroll
  for (int c = 0; c < 64; ++c) xn[pix * 64 + c] = vals[c] * inv * lnw[c];
}

// ---------------------------------------------------------------------------
// Kernel 2: QKV GEMM via WMMA f32 16x16x4.  C[pix][o] = sum_c XN[pix][c]*Wqkv[o][c]
// M=4096 (16px tiles), N=192 (12 tiles), K=64 (16 chunks of 4).
// Scatter: o<64 -> Qb[h][pix][d]; o<128 -> Kc[c][pix]; else Vc[c][pix].
// ---------------------------------------------------------------------------
__global__ __launch_bounds__(32) void qkv_gemm(const float* __restrict__ xn,
                                               const float* __restrict__ wqkv,
                                               float* __restrict__ Qb,
                                               float* __restrict__ Kc,
                                               float* __restrict__ Vc) {
  int l = threadIdx.x;
  int bid = blockIdx.x;      // 3072 tiles
  int mt = bid & 255;        // pixel tile
  int nt = bid >> 8;         // channel tile (0..11)
  int mbase = mt * 16, nbase = nt * 16;
  int lr = l & 15, hi = l >> 4;

  const float* arow = xn + (mbase + lr) * 64 + hi * 2;    // A: XN rows
  const float* brow = wqkv + (nbase + lr) * 64 + hi * 2;  // B: Wqkv^T cols

  v8f acc = {};
#pragma unroll
  for (int kc = 0; kc < 16; ++kc) {
    v2f a = *(const v2f*)(arow + kc * 4);
    v2f b = *(const v2f*)(brow + kc * 4);
    acc = wmma4(a, b, acc);
  }

  int o = nbase + lr;        // output channel (C-layout column = lane%16)
  int pixb = mbase + hi * 8; // C-layout row = pixel (g + 8*hi)
#pragma unroll
  for (int g = 0; g < 8; ++g) {
    int pix = pixb + g;
    float v = acc[g];
    if (o < 64) {
      int h = o >> 4, d = o & 15;
      Qb[(h * 4096 + pix) * 16 + d] = v;
    } else if (o < 128) {
      Kc[(o - 64) * 4096 + pix] = v;
    } else {
      Vc[(o - 128) * 4096 + pix] = v;
    }
  }
}

// ---------------------------------------------------------------------------
// Kernel 3: unfold gather reproducing the reference reshape exactly.
// KU[h][m][d] (row-major, m=9216,d=16); VUT[h][d][m] (transposed for A-operand).
// f = m*16+d ; c = h*16 + f/9216 ; r = f%9216 ; r = oi*768 + oj*64 + nh*8 + nw
// row = nh*8+oi-2 ; col = nw*8+oj-2 ; zero-pad OOB.
// ---------------------------------------------------------------------------
__global__ __launch_bounds__(256) void unfold_kernel(const float* __restrict__ Kc,
                                                     const float* __restrict__ Vc,
                                                     float* __restrict__ KU,
                                                     float* __restrict__ VUT) {
  int t = blockIdx.x * 256 + threadIdx.x;  // 589824 = 4*9216*16
  int d = t & 15;
  int m = (t >> 4) % 9216;
  int h = t / (9216 * 16);
  int f = m * 16 + d;
  int c = h * 16 + f / 9216;
  int r = f % 9216;
  int oi = r / 768;
  int oj = (r / 64) % 12;
  int win = r & 63;
  int row = (win >> 3) * 8 + oi - 2;
  int col = (win & 7) * 8 + oj - 2;
  bool ok = (row >= 0) && (row < 64) && (col >= 0) && (col < 64);
  float kv = ok ? Kc[c * 4096 + row * 64 + col] : 0.f;
  float vv = ok ? Vc[c * 4096 + row * 64 + col] : 0.f;
  KU[(h * 9216 + m) * 16 + d] = kv;
  VUT[(h * 16 + d) * 9216 + m] = vv;
}

// ---------------------------------------------------------------------------
// Kernel 4: flash attention. One wave = one (head, 16-query tile).
// Inner loop over 576 key tiles: S^T = K_tile x Q^T (4 wmma), online softmax
// (per-query state in lanes, halves merged with SWAPX16 swizzle),
// O^T += V^T x P^T (4 wmma) with P^T staged through wave-private LDS.
// ---------------------------------------------------------------------------
__global__ __launch_bounds__(32) void attn_kernel(const float* __restrict__ Qb,
                                                  const float* __restrict__ KU,
                                                  const float* __restrict__ VUT,
                                                  float* __restrict__ AO) {
  __shared__ float pbuf[256];  // P^T column-major: pbuf[q*16 + j]
  int l = threadIdx.x;
  int bid = blockIdx.x;  // 1024 = 4 heads * 256 query tiles
  int h = bid >> 8;
  int qbase = (bid & 255) * 16;
  int lr = l & 15, hi = l >> 4;

  // Q fragments, reused as B operand (B-layout of Q^T == A-layout of Q)
  const float* qrow = Qb + (h * 4096 + qbase + lr) * 16 + hi * 2;
  v2f qa[4];
#pragma unroll
  for (int kc = 0; kc < 4; ++kc) qa[kc] = *(const v2f*)(qrow + kc * 4);

  const float* kbase = KU + (h * 9216 + lr) * 16 + hi * 2;       // + mt*256 + kc*4
  const float* vbase = VUT + (h * 16 + lr) * 9216 + hi * 2;      // + mt*16  + kc*4

  float mrun = -3.0e38f;
  float lrun = 0.f;
  v8f o = {};

  for (int mt = 0; mt < 576; ++mt) {
    // S^T = K_tile(16x16) x Q^T(16x16), K-dim = DH = 16 in 4 chunks
    const float* kp = kbase + mt * 256;
    v8f st = {};
#pragma unroll
    for (int kc = 0; kc < 4; ++kc) {
      v2f ka = *(const v2f*)(kp + kc * 4);
      st = wmma4(ka, qa[kc], st);
    }
    // scale by DH^-0.5 = 0.25, online softmax over keys (rows of S^T)
    float mx = -3.0e38f;
#pragma unroll
    for (int g = 0; g < 8; ++g) {
      st[g] *= 0.25f;
      mx = fmaxf(mx, st[g]);
    }
    mx = fmaxf(mx, swap16(mx));           // merge the two key-halves
    float mnew = fmaxf(mrun, mx);
    float alpha = __expf(mrun - mnew);
    float p[8];
    float psum = 0.f;
#pragma unroll
    for (int g = 0; g < 8; ++g) {
      p[g] = __expf(st[g] - mnew);
      psum += p[g];
    }
    psum += swap16(psum);
    lrun = lrun * alpha + psum;
    mrun = mnew;
#pragma unroll
    for (int g = 0; g < 8; ++g) o[g] *= alpha;

    // stage P^T column-major: p[g] -> pbuf[lr*16 + g + 8*hi] (contiguous per half)
    {
      float* dst = &pbuf[lr * 16 + hi * 8];
      *(float4*)dst = make_float4(p[0], p[1], p[2], p[3]);
      *(float4*)(dst + 4) = make_float4(p[4], p[5], p[6], p[7]);
    }
    asm volatile("s_wait_dscnt 0x0" ::: "memory");  // LDS in-order per wave

    // O^T += V^T_chunk (A) x P^T_chunk (B)
    const float* vp = vbase + mt * 16;
#pragma unroll
    for (int kc = 0; kc < 4; ++kc) {
      v2f pb = *(const v2f*)(&pbuf[lr * 16 + kc * 4 + hi * 2]);
      v2f va = *(const v2f*)(vp + kc * 4);
      o = wmma4(va, pb, o);
    }
    asm volatile("" ::: "memory");  // keep next-iter LDS stores below the loads
  }

  // normalize and write AO[pix][h*16 + d]; O^T rows d = g + 8*hi, col q = lr
  float linv = 1.f / lrun;
#pragma unroll
  for (int g = 0; g < 8; ++g) o[g] *= linv;
  float* ao = AO + (qbase + lr) * 64 + h * 16 + hi * 8;
  *(float4*)ao = make_float4(o[0], o[1], o[2], o[3]);
  *(float4*)(ao + 4) = make_float4(o[4], o[5], o[6], o[7]);
}

// ---------------------------------------------------------------------------
// Kernel 5: output projection via WMMA. out[o][pix] = sum_c AO[pix][c]*Wout[o][c]
// ---------------------------------------------------------------------------
__global__ __launch_bounds__(32) void out_gemm(const float* __restrict__ AO,
                                               const float* __restrict__ wout,
                                               float* __restrict__ out) {
  int l = threadIdx.x;
  int bid = blockIdx.x;  // 1024 = 256 pixel tiles * 4 channel tiles
  int mt = bid & 255;
  int nt = bid >> 8;
  int mbase = mt * 16, nbase = nt * 16;
  int lr = l & 15, hi = l >> 4;

  const float* arow = AO + (mbase + lr) * 64 + hi * 2;
  const float* brow = wout + (nbase + lr) * 64 + hi * 2;
  v8f acc = {};
#pragma unroll
  for (int kc = 0; kc < 16; ++kc) {
    v2f a = *(const v2f*)(arow + kc * 4);
    v2f b = *(const v2f*)(brow + kc * 4);
    acc = wmma4(a, b, acc);
  }
  int o = nbase + lr;                 // output channel
  float* dst = out + o * 4096 + mbase + hi * 8;  // pixels contiguous over g
  *(float4*)dst = make_float4(acc[0], acc[1], acc[2], acc[3]);
  *(float4*)(dst + 4) = make_float4(acc[4], acc[5], acc[6], acc[7]);
}

// ---------------------------------------------------------------------------
extern "C" void kernel_launch(void* const* d_in, const int* in_sizes, int n_in,
                              void* d_out, int out_size, void* d_ws, size_t ws_size,
                              hipStream_t stream) {
  const float* x    = (const float*)d_in[0];   // (1,64,64,64)
  const float* wqkv = (const float*)d_in[1];   // (192,64)
  const float* wout = (const float*)d_in[2];   // (64,64)
  const float* lnw  = (const float*)d_in[3];   // (64,)

  float* ws  = (float*)d_ws;
  float* XN  = ws;                 // 4096*64        = 262144
  float* Qb  = ws + 262144;        // 4*4096*16      = 262144
  float* Kc  = ws + 524288;        // 64*4096        = 262144
  float* Vc  = ws + 786432;        // 64*4096        = 262144
  float* KU  = ws + 1048576;       // 4*9216*16      = 589824
  float* VUT = ws + 1638400;       // 4*16*9216      = 589824
  float* AO  = ws + 2228224;       // 4096*64        = 262144
  float* out = (float*)d_out;      // (64,4096)

  norm_kernel<<<16, 256, 0, stream>>>(x, lnw, XN);
  qkv_gemm<<<3072, 32, 0, stream>>>(XN, wqkv, Qb, Kc, Vc);
  unfold_kernel<<<2304, 256, 0, stream>>>(Kc, Vc, KU, VUT);
  attn_kernel<<<1024, 32, 0, stream>>>(Qb, KU, VUT, AO);
  out_gemm<<<1024, 32, 0, stream>>>(AO, wout, out);
}